// Net_11390253269714
// MI455X (gfx1250) — compile-verified
//
#include <hip/hip_runtime.h>
#include <hip/hip_bf16.h>

typedef __attribute__((ext_vector_type(2))) float v2f;
typedef __attribute__((ext_vector_type(8))) float v8f;

#define N_FEAT 32

// ---------------- degree / normalization ----------------

__global__ __launch_bounds__(256) void set_ones_k(float* __restrict__ deg, int n) {
    int i = blockIdx.x * blockDim.x + threadIdx.x;
    if (i < n) deg[i] = 1.0f;  // self-loop contributes 1 to every node's degree
}

__global__ __launch_bounds__(256) void deg_edges_k(const int* __restrict__ dst,
                                                   float* __restrict__ deg, int nE) {
    int e = blockIdx.x * blockDim.x + threadIdx.x;
    if (e < nE) atomicAdd(&deg[dst[e]], 1.0f);
}

__global__ __launch_bounds__(256) void rsqrt_k(float* __restrict__ deg, int n) {
    int i = blockIdx.x * blockDim.x + threadIdx.x;
    if (i < n) {
        float d = deg[i];
        deg[i] = (d > 0.0f) ? rsqrtf(d) : 0.0f;  // becomes dinv in place
    }
}

// ---------------- fc1: h0 = relu(x @ W1 + b1), x is [N,4], K=4 exactly ----------------

__global__ __launch_bounds__(256) void fc1_wmma_k(const float* __restrict__ x,
                                                  const float* __restrict__ W,   // [4,32]
                                                  const float* __restrict__ bias,// [32]
                                                  float* __restrict__ h,         // [N,32]
                                                  int n) {
    int wave    = (blockIdx.x * blockDim.x + threadIdx.x) >> 5;  // uniform per wave
    int lane    = threadIdx.x & 31;
    int half    = lane >> 4;
    int m       = lane & 15;
    int rowbase = wave * 16;
    if (rowbase >= n) return;  // wave-uniform exit: EXEC stays all-1s for WMMA

    int row = rowbase + m;
    if (row >= n) row = n - 1;

    // A: 16x4 f32, lane halves hold K={0,1} / K={2,3}
    v2f a;
    a.x = x[row * 4 + half * 2 + 0];
    a.y = x[row * 4 + half * 2 + 1];

    // B: 4x16 tiles of W (cols 0-15 and 16-31)
    int k = half * 2;
    v2f b0, b1;
    b0.x = W[(k + 0) * 32 + m];        b0.y = W[(k + 1) * 32 + m];
    b1.x = W[(k + 0) * 32 + 16 + m];   b1.y = W[(k + 1) * 32 + 16 + m];

    v8f c0 = {}, c1 = {};
    c0 = __builtin_amdgcn_wmma_f32_16x16x4_f32(false, a, false, b0, (short)0, c0, false, false);
    c1 = __builtin_amdgcn_wmma_f32_16x16x4_f32(false, a, false, b1, (short)0, c1, false, false);

    float bb0 = bias[m], bb1 = bias[16 + m];
    if (rowbase + 16 <= n) {
        // full tile: straight-line stores, no per-row EXEC churn
#pragma unroll
        for (int r = 0; r < 8; ++r) {
            int orow = rowbase + r + half * 8;
            h[orow * N_FEAT + m]      = fmaxf(c0[r] + bb0, 0.0f);
            h[orow * N_FEAT + 16 + m] = fmaxf(c1[r] + bb1, 0.0f);
        }
    } else {
#pragma unroll
        for (int r = 0; r < 8; ++r) {
            int orow = rowbase + r + half * 8;
            if (orow < n) {
                h[orow * N_FEAT + m]      = fmaxf(c0[r] + bb0, 0.0f);
                h[orow * N_FEAT + 16 + m] = fmaxf(c1[r] + bb1, 0.0f);
            }
        }
    }
}

// ---------------- conv projection: t = relu(in + bias_prev) @ W ; acc = t * dinv^2 ----------------

__global__ __launch_bounds__(256) void gemm32_wmma_k(const float* __restrict__ in,   // [N,32]
                                                     const float* __restrict__ bias, // [32] or null
                                                     const float* __restrict__ W,    // [32,32]
                                                     const float* __restrict__ dinv, // [N]
                                                     float* __restrict__ t,          // [N,32]
                                                     float* __restrict__ acc,        // [N,32]
                                                     int n) {
    int wave    = (blockIdx.x * blockDim.x + threadIdx.x) >> 5;
    int lane    = threadIdx.x & 31;
    int half    = lane >> 4;
    int m       = lane & 15;
    int rowbase = wave * 16;
    if (rowbase >= n) return;  // wave-uniform

    int row = rowbase + m;
    if (row >= n) row = n - 1;

    v8f c0 = {}, c1 = {};
#pragma unroll
    for (int kt = 0; kt < 8; ++kt) {
        int k = kt * 4 + half * 2;
        float x0 = in[row * N_FEAT + k];
        float x1 = in[row * N_FEAT + k + 1];
        if (bias) {  // fused bias+ReLU of the previous conv layer (uniform branch)
            x0 = fmaxf(x0 + bias[k], 0.0f);
            x1 = fmaxf(x1 + bias[k + 1], 0.0f);
        }
        v2f a; a.x = x0; a.y = x1;
        v2f b0, b1;
        b0.x = W[(k + 0) * 32 + m];       b0.y = W[(k + 1) * 32 + m];
        b1.x = W[(k + 0) * 32 + 16 + m];  b1.y = W[(k + 1) * 32 + 16 + m];
        c0 = __builtin_amdgcn_wmma_f32_16x16x4_f32(false, a, false, b0, (short)0, c0, false, false);
        c1 = __builtin_amdgcn_wmma_f32_16x16x4_f32(false, a, false, b1, (short)0, c1, false, false);
    }

    if (rowbase + 16 <= n) {
        // full tile: unpredicated store burst
#pragma unroll
        for (int r = 0; r < 8; ++r) {
            int orow = rowbase + r + half * 8;
            float di = dinv[orow];
            float s  = di * di;  // self-loop norm
            float v0 = c0[r], v1 = c1[r];
            t[orow * N_FEAT + m]        = v0;
            t[orow * N_FEAT + 16 + m]   = v1;
            acc[orow * N_FEAT + m]      = v0 * s;
            acc[orow * N_FEAT + 16 + m] = v1 * s;
        }
    } else {
#pragma unroll
        for (int r = 0; r < 8; ++r) {
            int orow = rowbase + r + half * 8;
            if (orow < n) {
                float di = dinv[orow];
                float s  = di * di;
                float v0 = c0[r], v1 = c1[r];
                t[orow * N_FEAT + m]        = v0;
                t[orow * N_FEAT + 16 + m]   = v1;
                acc[orow * N_FEAT + m]      = v0 * s;
                acc[orow * N_FEAT + 16 + m] = v1 * s;
            }
        }
    }
}

// ---------------- edge message passing: acc[dst] += t[src] * dinv[src]*dinv[dst] ----------------
// One wave32 per edge; lane = feature column -> one coalesced 128B gather line and
// 32 consecutive-address f32 atomics per edge. Edge id forced wave-uniform so the
// index/dinv loads become scalar (SMEM) loads and addresses use saddr form.

__global__ __launch_bounds__(256) void edge_msg_k(const int* __restrict__ src,
                                                  const int* __restrict__ dst,
                                                  const float* __restrict__ dinv,
                                                  const float* __restrict__ t,
                                                  float* __restrict__ acc, int nE) {
    int e = __builtin_amdgcn_readfirstlane(
        blockIdx.x * (blockDim.x >> 5) + (threadIdx.x >> 5));
    int col = threadIdx.x & 31;
    if (e >= nE) return;  // uniform branch
    int s = __builtin_amdgcn_readfirstlane(src[e]);
    int d = __builtin_amdgcn_readfirstlane(dst[e]);
    float w = dinv[s] * dinv[d];
    float v = t[(size_t)s * N_FEAT + col] * w;
    atomicAdd(&acc[(size_t)d * N_FEAT + col], v);
}

// ---------------- fc2: out = relu(acc + b_c3) @ W_fc2 + b_fc2 ----------------

__global__ __launch_bounds__(256) void fc2_k(const float* __restrict__ acc,
                                             const float* __restrict__ bc3,  // [32]
                                             const float* __restrict__ W2,   // [32,3]
                                             const float* __restrict__ b2,   // [3]
                                             float* __restrict__ out, int n) {
    int i = blockIdx.x * blockDim.x + threadIdx.x;
    if (i >= n) return;
    float o0 = b2[0], o1 = b2[1], o2 = b2[2];
#pragma unroll
    for (int j = 0; j < 32; ++j) {
        float hj = fmaxf(acc[(size_t)i * N_FEAT + j] + bc3[j], 0.0f);
        o0 += hj * W2[j * 3 + 0];
        o1 += hj * W2[j * 3 + 1];
        o2 += hj * W2[j * 3 + 2];
    }
    out[i * 3 + 0] = o0;
    out[i * 3 + 1] = o1;
    out[i * 3 + 2] = o2;
}

// ---------------- host launcher ----------------

extern "C" void kernel_launch(void* const* d_in, const int* in_sizes, int n_in,
                              void* d_out, int out_size, void* d_ws, size_t ws_size,
                              hipStream_t stream) {
    const float* x    = (const float*)d_in[0];
    const int*   ei   = (const int*)  d_in[1];
    const float* Wf1  = (const float*)d_in[2];
    const float* bf1  = (const float*)d_in[3];
    const float* Wc1  = (const float*)d_in[4];
    const float* bc1  = (const float*)d_in[5];
    const float* Wc2  = (const float*)d_in[6];
    const float* bc2  = (const float*)d_in[7];
    const float* Wc3  = (const float*)d_in[8];
    const float* bc3  = (const float*)d_in[9];
    const float* Wf2  = (const float*)d_in[10];
    const float* bf2  = (const float*)d_in[11];
    float*       out  = (float*)d_out;

    const int N = in_sizes[0] / 4;   // 100000 nodes
    const int E = in_sizes[1] / 2;   // 2.5M edges
    const int* src = ei;
    const int* dst = ei + E;

    float* dinv = (float*)d_ws;                  // [N]
    float* bufA = dinv + N;                      // [N,32]
    float* bufB = bufA + (size_t)N * N_FEAT;     // [N,32]
    float* bufT = bufB + (size_t)N * N_FEAT;     // [N,32]

    const int TB = 256;
    int nBlk  = (N + TB - 1) / TB;
    int eBlk  = (E + TB - 1) / TB;
    int waves = (N + 15) / 16;
    int gBlk  = (waves * 32 + TB - 1) / TB;                 // WMMA kernels: 1 wave per 16 rows
    int mBlk  = (int)(((unsigned)E * 32u + TB - 1) / TB);   // edge kernel: 1 wave per edge

    // degree -> dinv (self-loops included via init to 1)
    set_ones_k <<<nBlk, TB, 0, stream>>>(dinv, N);
    deg_edges_k<<<eBlk, TB, 0, stream>>>(dst, dinv, E);
    rsqrt_k    <<<nBlk, TB, 0, stream>>>(dinv, N);

    // h0 = relu(x @ W_fc1 + b_fc1)  -> bufA
    fc1_wmma_k<<<gBlk, TB, 0, stream>>>(x, Wf1, bf1, bufA, N);

    // conv1: in=bufA (already relu'd, no bias fuse) -> t=bufT, acc=bufB
    gemm32_wmma_k<<<gBlk, TB, 0, stream>>>(bufA, nullptr, Wc1, dinv, bufT, bufB, N);
    edge_msg_k   <<<mBlk, TB, 0, stream>>>(src, dst, dinv, bufT, bufB, E);

    // conv2: in=relu(bufB + bc1) -> acc=bufA
    gemm32_wmma_k<<<gBlk, TB, 0, stream>>>(bufB, bc1, Wc2, dinv, bufT, bufA, N);
    edge_msg_k   <<<mBlk, TB, 0, stream>>>(src, dst, dinv, bufT, bufA, E);

    // conv3: in=relu(bufA + bc2) -> acc=bufB
    gemm32_wmma_k<<<gBlk, TB, 0, stream>>>(bufA, bc2, Wc3, dinv, bufT, bufB, N);
    edge_msg_k   <<<mBlk, TB, 0, stream>>>(src, dst, dinv, bufT, bufB, E);

    // out = relu(bufB + bc3) @ W_fc2 + b_fc2
    fc2_k<<<nBlk, TB, 0, stream>>>(bufB, bc3, Wf2, bf2, out, N);
}